// geomm_net1_38568806318241
// MI455X (gfx1250) — compile-verified
//
#include <hip/hip_runtime.h>
#include <hip/hip_bf16.h>

// Problem constants (from the reference)
#define NS     24
#define BB     16
#define CC     512
#define PLANE  256          // 16*16 spatial elements, contiguous per (.., c) plane
#define FEATK  1024         // concat(feat_f, feat_o) width
#define NCLS   3

typedef __attribute__((ext_vector_type(4))) float v4f;
typedef __attribute__((ext_vector_type(2))) float v2f;
typedef __attribute__((ext_vector_type(8))) float v8f;

// ---------------------------------------------------------------------------
// Kernel 1: streaming reduction. One wave32 per (b, c) pair.
//   feat[b, c]        = mean_{hw} feat_f_map[b, c, :, :]
//   feat[b, 512 + c]  = max_s mean_{hw} oct_maps[s, b, c, :, :]
// (the argmax+gather in the reference collapses to a max of plane means)
// Each 1KB plane is read as two fully-coalesced 512B wave transactions
// (lane L -> float4[L] and float4[L+32]) with non-temporal hints: 209MB
// streamed once, nothing reused -> bandwidth bound at ~9us on 23.3 TB/s.
// ---------------------------------------------------------------------------
__device__ __forceinline__ float plane_sum32(const float* __restrict__ p, int lane) {
    const v4f* p4 = reinterpret_cast<const v4f*>(p);
    v4f x0 = __builtin_nontemporal_load(&p4[lane]);
    v4f x1 = __builtin_nontemporal_load(&p4[lane + 32]);
    float s = (x0.x + x0.y) + (x0.z + x0.w) + (x1.x + x1.y) + (x1.z + x1.w);
    // wave32 butterfly reduction
    #pragma unroll
    for (int off = 16; off > 0; off >>= 1)
        s += __shfl_xor(s, off, 32);
    return s;
}

__global__ __launch_bounds__(256) void reduce_feat_kernel(
        const float* __restrict__ feat_f_map,   // (B, C, H, W)
        const float* __restrict__ oct_maps,     // (NS, B, C, H, W)
        float* __restrict__ feat)               // (B, 1024) row-major, in d_ws
{
    const int lane = threadIdx.x & 31;
    const int wave = blockIdx.x * (blockDim.x >> 5) + (threadIdx.x >> 5);
    const int b = wave >> 9;          // wave / 512
    const int c = wave & (CC - 1);    // wave % 512

    const float inv = 1.0f / (float)PLANE;

    float ff = plane_sum32(feat_f_map + ((size_t)b * CC + c) * PLANE, lane) * inv;

    float fo = -__builtin_inff();
    #pragma unroll 4
    for (int s = 0; s < NS; ++s) {
        float m = plane_sum32(oct_maps + (((size_t)s * BB + b) * CC + c) * PLANE, lane) * inv;
        fo = fmaxf(fo, m);
    }

    if (lane == 0) {
        feat[b * FEATK + c]      = ff;
        feat[b * FEATK + CC + c] = fo;
    }
}

// ---------------------------------------------------------------------------
// Kernel 2: head GEMM via V_WMMA_F32_16X16X4_F32, single wave32.
//   labels(16x3) = feat(16x1024) @ head_w(3x1024)^T + head_b
// N padded 3 -> 16 by masking B columns >= 3 to zero (multiply mask, no
// branch, so EXEC is all-1s at every WMMA as the ISA requires).
//
// A layout (16x4 f32): lanes 0-15 hold row M=lane, {K+0, K+1} in v[0],v[1];
//                      lanes 16-31 hold row M=lane-16, {K+2, K+3}.
// B layout mirrors A with N in place of M. C/D: VGPR r holds M = 8*half + r,
// N = lane % 16.
// ---------------------------------------------------------------------------
__global__ __launch_bounds__(32) void head_wmma_kernel(
        const float* __restrict__ feat,     // (16, 1024)
        const float* __restrict__ head_w,   // (3, 1024)
        const float* __restrict__ head_b,   // (3,)
        float* __restrict__ out)            // (16, 3)
{
    const int lane = threadIdx.x;       // 0..31
    const int half = lane >> 4;         // 0 or 1
    const int l16  = lane & 15;

    const int   n     = l16;                       // output class column
    const int   nc    = (n < NCLS) ? n : (NCLS - 1);
    const float bmask = (n < NCLS) ? 1.0f : 0.0f;  // zero padded columns

    const float* arow = feat   + (size_t)l16 * FEATK + 2 * half;  // A: row M=l16
    const float* brow = head_w + (size_t)nc  * FEATK + 2 * half;  // B: col n (=head_w row)

    v8f acc = {};
    for (int k0 = 0; k0 < FEATK; k0 += 4) {
        v2f a = *reinterpret_cast<const v2f*>(arow + k0);  // {K+koff, K+koff+1}
        v2f bm;
        bm.x = brow[k0]     * bmask;
        bm.y = brow[k0 + 1] * bmask;
        // 8 args: (neg_a, A, neg_b, B, c_mod, C, reuse_a, reuse_b)
        acc = __builtin_amdgcn_wmma_f32_16x16x4_f32(
                false, a, false, bm, (short)0, acc, false, false);
    }

    if (n < NCLS) {
        const float bias = head_b[n];
        #pragma unroll
        for (int r = 0; r < 8; ++r) {
            const int m = half * 8 + r;         // C/D row from VGPR layout
            out[m * NCLS + n] = acc[r] + bias;
        }
    }
}

// ---------------------------------------------------------------------------
extern "C" void kernel_launch(void* const* d_in, const int* in_sizes, int n_in,
                              void* d_out, int out_size, void* d_ws, size_t ws_size,
                              hipStream_t stream) {
    const float* feat_f_map = (const float*)d_in[0];  // (16, 512, 16, 16)
    const float* oct_maps   = (const float*)d_in[1];  // (24, 16, 512, 16, 16)
    const float* head_w     = (const float*)d_in[2];  // (3, 1024)
    const float* head_b     = (const float*)d_in[3];  // (3,)
    float*       out        = (float*)d_out;          // (16, 3)
    float*       feat       = (float*)d_ws;           // (16, 1024) scratch, 64 KB

    // 8192 (b,c) pairs, 8 waves per 256-thread block -> 1024 blocks.
    reduce_feat_kernel<<<1024, 256, 0, stream>>>(feat_f_map, oct_maps, feat);

    // Tiny head GEMM: one wave, fp32 WMMA.
    head_wmma_kernel<<<1, 32, 0, stream>>>(feat, head_w, head_b, out);
}